// COHESION_9921374454293
// MI455X (gfx1250) — compile-verified
//
#include <hip/hip_runtime.h>
#include <math.h>

#define NU  150000
#define NI  75000
#define DF  768
#define DL  64
#define HID 256
#define NTOTAL 225000
#define NNZ_ 3000000

typedef __attribute__((ext_vector_type(2))) float v2f;
typedef __attribute__((ext_vector_type(8))) float v8f;

// -------------------------------------------------------------------------
// WMMA fp32 GEMM:  out = EPI( A[M,KTOT] @ B[KTOT,NTOT] + bias )
//   EPI==0 : leaky_relu(.,0.01)           -> out[m*NTOT+n]         (GEMM1)
//   EPI==1 : sqrt(|(id^2+h^2)/2 + 1e-8|)  -> out[(NU+m)*DL+n]      (GEMM2)
// block = 128 threads (4 waves); each wave computes a 16x64 tile.
// A in LDS row-major [64][16] (pad 20); B in LDS TRANSPOSED [n][k] (pad 20)
// so each lane's A and B fragments are single aligned ds_load_b64's.
// -------------------------------------------------------------------------
template<int KTOT, int NTOT, int EPI>
__launch_bounds__(128)
__global__ void wmma_gemm_kernel(const float* __restrict__ A,
                                 const float* __restrict__ B,
                                 const float* __restrict__ bias,
                                 const float* __restrict__ aux,
                                 float* __restrict__ out, int M)
{
    __shared__ float As[64][20];   // As[m_local][k_local]
    __shared__ float Bt[64][20];   // Bt[n_local][k_local]  (transposed)

    const int tid  = threadIdx.x;
    const int lane = tid & 31;
    const int w    = tid >> 5;       // wave 0..3 -> rows [w*16, w*16+16)
    const int mb   = blockIdx.y * 64;
    const int nb   = blockIdx.x * 64;
    const int lm   = lane & 15;
    const int hh   = lane >> 4;      // half-wave select
    const int kk   = hh * 2;         // K-pair base within a K=4 step

    v8f acc[4];
    #pragma unroll
    for (int i = 0; i < 4; ++i)
        #pragma unroll
        for (int j = 0; j < 8; ++j) acc[i][j] = 0.0f;

    const int arow = tid >> 1, acol = (tid & 1) * 8;   // A tile 64x16
    const int bn   = tid & 63, bkq  = (tid >> 6) * 8;  // B tile 16x64, column order

    for (int kc = 0; kc < KTOT; kc += 16) {
        // ---- global loads for this chunk (before barrier: overlap w/ compute)
        float4 a0, a1;
        if (mb + arow < M) {
            const float* ap = A + (size_t)(mb + arow) * KTOT + kc + acol;
            a0 = *(const float4*)(ap);
            a1 = *(const float4*)(ap + 4);
        } else {
            a0 = make_float4(0.f, 0.f, 0.f, 0.f);
            a1 = make_float4(0.f, 0.f, 0.f, 0.f);
        }
        // column fragment of B: 8 coalesced b32 loads (64 lanes x 4B rows)
        float breg[8];
        #pragma unroll
        for (int j = 0; j < 8; ++j)
            breg[j] = B[(size_t)(kc + bkq + j) * NTOT + nb + bn];

        __syncthreads();                       // previous tile fully consumed
        *(float4*)&As[arow][acol]     = a0;
        *(float4*)&As[arow][acol + 4] = a1;
        *(float4*)&Bt[bn][bkq]     = make_float4(breg[0], breg[1], breg[2], breg[3]);
        *(float4*)&Bt[bn][bkq + 4] = make_float4(breg[4], breg[5], breg[6], breg[7]);
        __syncthreads();

        #pragma unroll
        for (int ks = 0; ks < 4; ++ks) {       // 4 x (K=4) per chunk
            const int kb = ks * 4 + kk;
            const v2f a = *(const v2f*)&As[w * 16 + lm][kb];
            #pragma unroll
            for (int nt = 0; nt < 4; ++nt) {
                const v2f b = *(const v2f*)&Bt[nt * 16 + lm][kb];
                acc[nt] = __builtin_amdgcn_wmma_f32_16x16x4_f32(
                    false, a, false, b, (short)0, acc[nt], false, false);
            }
        }
    }

    #pragma unroll
    for (int nt = 0; nt < 4; ++nt) {
        const int n  = nb + nt * 16 + lm;
        const float bv = bias[n];
        #pragma unroll
        for (int v = 0; v < 8; ++v) {
            const int m = mb + w * 16 + v + 8 * hh;
            if (m < M) {
                float h = acc[nt][v] + bv;
                if (EPI == 0) {
                    h = h > 0.0f ? h : 0.01f * h;
                    out[(size_t)m * NTOT + n] = h;
                } else {
                    const float idv = aux[(size_t)m * DL + n];
                    const float t = sqrtf(fabsf((idv * idv + h * h) * 0.5f + 1e-8f));
                    out[(size_t)(NU + m) * DL + n] = t;
                }
            }
        }
    }
}

// Row L2-normalize: users from `pref`, items in-place from x (temp already there).
// Writes normalized row to x and initializes total(=d_out) with it.
__launch_bounds__(256)
__global__ void normalize_init_kernel(const float* __restrict__ pref,
                                      float* __restrict__ x,
                                      float* __restrict__ total)
{
    const int row = blockIdx.x * 8 + (threadIdx.x >> 5);
    if (row >= NTOTAL) return;
    const int lane = threadIdx.x & 31;
    const float* src = (row < NU) ? (pref + (size_t)row * DL) : (x + (size_t)row * DL);
    float v0 = src[lane], v1 = src[lane + 32];
    float ss = v0 * v0 + v1 * v1;
    #pragma unroll
    for (int off = 16; off > 0; off >>= 1) ss += __shfl_xor(ss, off, 32);
    const float inv = 1.0f / fmaxf(sqrtf(ss), 1e-12f);
    v0 *= inv; v1 *= inv;
    float* xr = x + (size_t)row * DL;
    float* tr = total + (size_t)row * DL;
    xr[lane] = v0; xr[lane + 32] = v1;
    tr[lane] = v0; tr[lane + 32] = v1;
}

__global__ void zero_kernel(float4* __restrict__ p, long n4)
{
    const long i = (long)blockIdx.x * blockDim.x + threadIdx.x;
    if (i < n4) p[i] = make_float4(0.f, 0.f, 0.f, 0.f);
}

// One wave per COO edge: dst[row] += val * src[col]  (64 f32 via 2 scalar f32 atomics per lane).
// Dense operand (57.6 MB) fits in the 192 MB L2, so gathers/atomics stay on-chip.
__launch_bounds__(256)
__global__ void spmm_kernel(const int* __restrict__ rows, const int* __restrict__ cols,
                            const float* __restrict__ vals,
                            const float* __restrict__ src, float* __restrict__ dst)
{
    const int e = blockIdx.x * 8 + (threadIdx.x >> 5);
    if (e >= NNZ_) return;
    const int lane = threadIdx.x & 31;
    const int r = rows[e], c = cols[e];
    const float v = vals[e];
    const float2 s = *(const float2*)(src + (size_t)c * DL + lane * 2);
    float* d = dst + (size_t)r * DL + lane * 2;
    atomicAdd(d,     v * s.x);
    atomicAdd(d + 1, v * s.y);
}

// Cosine reweight against ego(=x) + accumulate into total. One wave per row.
__launch_bounds__(256)
__global__ void reweight_kernel(float* __restrict__ e, const float* __restrict__ ego,
                                float* __restrict__ total)
{
    const int row = blockIdx.x * 8 + (threadIdx.x >> 5);
    if (row >= NTOTAL) return;
    const int lane = threadIdx.x & 31;
    float* er = e + (size_t)row * DL;
    const float* gr = ego + (size_t)row * DL;
    float e0 = er[lane], e1 = er[lane + 32];
    const float g0 = gr[lane], g1 = gr[lane + 32];
    float see = e0 * e0 + e1 * e1;
    float seg = e0 * g0 + e1 * g1;
    float sgg = g0 * g0 + g1 * g1;
    #pragma unroll
    for (int off = 16; off > 0; off >>= 1) {
        see += __shfl_xor(see, off, 32);
        seg += __shfl_xor(seg, off, 32);
        sgg += __shfl_xor(sgg, off, 32);
    }
    const float wgt = seg / (fmaxf(sqrtf(see), 1e-8f) * fmaxf(sqrtf(sgg), 1e-8f));
    e0 *= wgt; e1 *= wgt;
    er[lane] = e0; er[lane + 32] = e1;
    float* tr = total + (size_t)row * DL;
    tr[lane]      += e0;
    tr[lane + 32] += e1;
}

extern "C" void kernel_launch(void* const* d_in, const int* in_sizes, int n_in,
                              void* d_out, int out_size, void* d_ws, size_t ws_size,
                              hipStream_t stream)
{
    const float* features = (const float*)d_in[0];
    const float* id_embd  = (const float*)d_in[1];
    const float* pref     = (const float*)d_in[2];
    const float* W1 = (const float*)d_in[3];
    const float* b1 = (const float*)d_in[4];
    const float* W2 = (const float*)d_in[5];
    const float* b2 = (const float*)d_in[6];
    const int*   adj   = (const int*)d_in[7];
    const float* avals = (const float*)d_in[8];
    (void)in_sizes; (void)n_in; (void)out_size; (void)ws_size;

    float* total = (float*)d_out;                 // total accumulator == output
    float* ws    = (float*)d_ws;
    float* x     = ws;                            // [N, 64]  ego embeddings
    float* bufA  = ws + (size_t)NTOTAL * DL;      // ping
    float* bufB  = bufA + (size_t)NTOTAL * DL;    // pong
    float* h1    = bufA;                          // [NI,256] aliases ping+pong (dead before layers)

    const int* rows = adj;
    const int* cols = adj + NNZ_;

    // MLP via fp32 WMMA
    wmma_gemm_kernel<DF, HID, 0><<<dim3(HID / 64, (NI + 63) / 64), 128, 0, stream>>>(
        features, W1, b1, nullptr, h1, NI);
    wmma_gemm_kernel<HID, DL, 1><<<dim3(1, (NI + 63) / 64), 128, 0, stream>>>(
        h1, W2, b2, id_embd, x, NI);

    // concat + F.normalize + total = x
    normalize_init_kernel<<<(NTOTAL + 7) / 8, 256, 0, stream>>>(pref, x, total);

    // 3 propagation layers (num_layer fixed at 3 in setup)
    const float* src = x;
    float* bufs[2] = {bufA, bufB};
    const long n4 = (long)NTOTAL * DL / 4;
    for (int l = 0; l < 3; ++l) {
        float* dst = bufs[l & 1];
        zero_kernel<<<(unsigned)((n4 + 255) / 256), 256, 0, stream>>>((float4*)dst, n4);
        spmm_kernel<<<(NNZ_ + 7) / 8, 256, 0, stream>>>(rows, cols, avals, src, dst);
        reweight_kernel<<<(NTOTAL + 7) / 8, 256, 0, stream>>>(dst, x, total);
        src = dst;
    }
}